// OptLayer_3977139716219
// MI455X (gfx1250) — compile-verified
//
#include <hip/hip_runtime.h>
#include <stdint.h>

typedef __attribute__((ext_vector_type(2))) float v2f;
typedef __attribute__((ext_vector_type(8))) float v8f;

#define B_ROWS   4096
#define K_DIM    2048
#define N_DIM    1024
#define M_TILE   16
#define KC       64
#define NCHUNK   (K_DIM / KC)
#define AS_STRIDE 68   // 64 + 4 pad: kills LDS bank conflicts on A-fragment b64 reads

// ---------------------------------------------------------------------------
// Kernel 1: z = x * W^T - b   via V_WMMA_F32_16X16X4_F32 (exact fp32 path)
// grid: 256 blocks (16 rows each), block: 256 threads = 8 waves (wave32)
// wave w owns columns [w*128, w*128+128) -> 8 accumulator tiles of 16x16
// A tile staged by GLOBAL_LOAD_ASYNC_TO_LDS_B128, double-buffered, ASYNCcnt.
// ---------------------------------------------------------------------------
__global__ __launch_bounds__(256)
void gemm_bias_kernel(const float* __restrict__ x,
                      const float* __restrict__ W,
                      const float* __restrict__ bias,
                      float* __restrict__ z)
{
    __shared__ float As[2][M_TILE * AS_STRIDE];

    const int tid   = threadIdx.x;
    const int lane  = tid & 31;
    const int wave  = tid >> 5;
    const int r0    = blockIdx.x * M_TILE;
    const int cn0   = wave * 128;
    const int lm    = lane & 15;          // M index (A) / N index (B,C)
    const int khalf = (lane >> 4) * 2;    // lanes 0-15 hold K={0,1}, 16-31 hold K={2,3}

    v8f acc[8];
    #pragma unroll
    for (int nt = 0; nt < 8; ++nt)
        acc[nt] = (v8f){0.f,0.f,0.f,0.f,0.f,0.f,0.f,0.f};

    // cooperative A-tile mapping: 256 threads x one b128 = 16x64 fp32 tile
    const int arow = tid >> 4;            // 0..15
    const int acol = (tid & 15) * 4;      // 0..60
    const float*   gsrc     = &x[(size_t)(r0 + arow) * K_DIM + acol];
    const uint32_t lds_off0 = (uint32_t)(uintptr_t)&As[0][arow * AS_STRIDE + acol];
    const uint32_t lds_off1 = (uint32_t)(uintptr_t)&As[1][arow * AS_STRIDE + acol];

    // prologue: chunk 0 -> buffer 0 (async DMA into LDS, tracked by ASYNCcnt)
    asm volatile("global_load_async_to_lds_b128 %0, %1, off"
                 :: "v"(lds_off0), "v"(gsrc) : "memory");

    for (int c = 0; c < NCHUNK; ++c) {
        // my async fill of the current buffer is done, then rendezvous
        asm volatile("s_wait_asynccnt 0x0" ::: "memory");
        __syncthreads();

        // kick off next chunk into the other buffer (everyone is already past
        // the compute that read it, thanks to the barrier above)
        if (c + 1 < NCHUNK) {
            const float*   gn   = gsrc + (c + 1) * KC;
            const uint32_t ldst = ((c + 1) & 1) ? lds_off1 : lds_off0;
            asm volatile("global_load_async_to_lds_b128 %0, %1, off"
                         :: "v"(ldst), "v"(gn) : "memory");
        }

        const float* Ab = As[c & 1];
        const int    kc = c * KC;

        #pragma unroll
        for (int j = 0; j < 16; ++j) {
            const int kk = j * 4;
            // A fragment: 16x4 fp32, one row per lane group, b64 from LDS
            v2f afrag = *(const v2f*)&Ab[lm * AS_STRIDE + kk + khalf];
            #pragma unroll
            for (int nt = 0; nt < 8; ++nt) {
                const int n = cn0 + nt * 16 + lm;      // output column = W row
                // B fragment: B[k][n] = W[n][k]; W is L2-resident (8 MB << 192 MB)
                v2f bfrag = *(const v2f*)&W[(size_t)n * K_DIM + kc + kk + khalf];
                acc[nt] = __builtin_amdgcn_wmma_f32_16x16x4_f32(
                    false, afrag, false, bfrag, (short)0, acc[nt], false, false);
            }
        }
    }

    // C/D layout: VGPR v, lanes 0-15 -> M=v, lanes 16-31 -> M=v+8, N=lane%16
    const int rbase = (lane >> 4) * 8;
    #pragma unroll
    for (int nt = 0; nt < 8; ++nt) {
        const int col = cn0 + nt * 16 + lm;
        const float bv = bias[col];
        #pragma unroll
        for (int v = 0; v < 8; ++v) {
            z[(size_t)(r0 + rbase + v) * N_DIM + col] = acc[nt][v] - bv;
        }
    }
}

// ---------------------------------------------------------------------------
// 16-lane-group reductions (row group lives inside one half of a wave32)
// ---------------------------------------------------------------------------
__device__ __forceinline__ float rsum16(float v) {
    #pragma unroll
    for (int m = 1; m < 16; m <<= 1) v += __shfl_xor(v, m, 32);
    return v;
}
__device__ __forceinline__ float rmin16(float v) {
    #pragma unroll
    for (int m = 1; m < 16; m <<= 1) v = fminf(v, __shfl_xor(v, m, 32));
    return v;
}
__device__ __forceinline__ float rmax16(float v) {
    #pragma unroll
    for (int m = 1; m < 16; m <<= 1) v = fmaxf(v, __shfl_xor(v, m, 32));
    return v;
}

// ---------------------------------------------------------------------------
// Kernel 2: register-resident capped-simplex projection, 100 bisection iters
// 16 lanes per row, 64 elements (16 float4) per lane kept in VGPRs.
// Reference streams z from HBM 100x (~3.2 GB); we read it once (L2-resident).
// ---------------------------------------------------------------------------
__global__ __launch_bounds__(256)
void proj_kernel(const float* __restrict__ z,
                 const float* __restrict__ u,
                 float* __restrict__ y)
{
    const int tid = threadIdx.x;
    const int sub = tid & 15;                       // lane within row group
    const int row = blockIdx.x * 16 + (tid >> 4);

    float4 zv[16], uv[16];
    float zmin =  3.4e38f, zmax = -3.4e38f, umax = -3.4e38f;

    #pragma unroll
    for (int jj = 0; jj < 16; ++jj) {
        const int col = jj * 64 + sub * 4;          // coalesced 256B per half-wave
        zv[jj] = *(const float4*)&z[(size_t)row * N_DIM + col];
        uv[jj] = *(const float4*)&u[col];
        zmin = fminf(zmin, fminf(fminf(zv[jj].x, zv[jj].y), fminf(zv[jj].z, zv[jj].w)));
        zmax = fmaxf(zmax, fmaxf(fmaxf(zv[jj].x, zv[jj].y), fmaxf(zv[jj].z, zv[jj].w)));
        umax = fmaxf(umax, fmaxf(fmaxf(uv[jj].x, uv[jj].y), fmaxf(uv[jj].z, uv[jj].w)));
    }
    zmin = rmin16(zmin);
    zmax = rmax16(zmax);
    umax = rmax16(umax);

    // matches reference: lo = min(z) - max(u), hi = max(z)
    float lo = zmin - umax;
    float hi = zmax;

    for (int it = 0; it < 100; ++it) {
        const float mid = 0.5f * (lo + hi);
        float s = 0.0f;
        #pragma unroll
        for (int jj = 0; jj < 16; ++jj) {
            s += fminf(fmaxf(zv[jj].x - mid, 0.0f), uv[jj].x);
            s += fminf(fmaxf(zv[jj].y - mid, 0.0f), uv[jj].y);
            s += fminf(fmaxf(zv[jj].z - mid, 0.0f), uv[jj].z);
            s += fminf(fmaxf(zv[jj].w - mid, 0.0f), uv[jj].w);
        }
        s = rsum16(s);
        const bool big = (s > 90.0f);
        lo = big ? mid : lo;    // branchless; identical across the 16 lanes
        hi = big ? hi  : mid;
    }

    const float tau = 0.5f * (lo + hi);
    #pragma unroll
    for (int jj = 0; jj < 16; ++jj) {
        const int col = jj * 64 + sub * 4;
        float4 o;
        o.x = fminf(fmaxf(zv[jj].x - tau, 0.0f), uv[jj].x);
        o.y = fminf(fmaxf(zv[jj].y - tau, 0.0f), uv[jj].y);
        o.z = fminf(fmaxf(zv[jj].z - tau, 0.0f), uv[jj].z);
        o.w = fminf(fmaxf(zv[jj].w - tau, 0.0f), uv[jj].w);
        *(float4*)&y[(size_t)row * N_DIM + col] = o;
    }
}

// ---------------------------------------------------------------------------
extern "C" void kernel_launch(void* const* d_in, const int* in_sizes, int n_in,
                              void* d_out, int out_size, void* d_ws, size_t ws_size,
                              hipStream_t stream)
{
    const float* x  = (const float*)d_in[0];   // [4096, 2048]
    const float* W  = (const float*)d_in[1];   // [1024, 2048]
    const float* b  = (const float*)d_in[2];   // [1024]
    const float* u  = (const float*)d_in[3];   // [1024]
    float*       y  = (float*)d_out;           // [4096, 1024]
    float*       zw = (float*)d_ws;            // scratch z: 16 MB

    gemm_bias_kernel<<<B_ROWS / M_TILE, 256, 0, stream>>>(x, W, b, zw);
    proj_kernel<<<B_ROWS / 16, 256, 0, stream>>>(zw, u, y);
}